// Biholomorphic_k2_45921790329588
// MI455X (gfx1250) — compile-verified
//
#include <hip/hip_runtime.h>
#include <hip/hip_bf16.h>

// CDNA5 / gfx1250: wave32, WMMA 16x16x4 F32
typedef __attribute__((ext_vector_type(2))) float v2f;
typedef __attribute__((ext_vector_type(8))) float v8f;

#define DIM5            5
#define NZZ             15
#define NOUT            225
#define WAVES_PER_BLOCK 8
#define THREADS         (WAVES_PER_BLOCK * 32)
#define GRID_BLOCKS     4096

// ---- compile-time tables -------------------------------------------------
// triu_indices(5): 15 (i,j) pairs, row-major, packed 3 bits per entry
constexpr int I5T[NZZ] = {0,0,0,0,0,1,1,1,1,2,2,2,3,3,4};
constexpr int J5T[NZZ] = {0,1,2,3,4,1,2,3,4,2,3,4,3,4,4};

constexpr unsigned long long pack3(const int (&t)[NZZ]) {
    unsigned long long v = 0;
    for (int k = 0; k < NZZ; ++k) v |= (unsigned long long)(t[k] & 7) << (3 * k);
    return v;
}

// LDS gather offsets (float index in the wave's 512-float region):
//   k <  120 : real matrix (base 0),   entry (i,j), i<=j, row-major
//   k >= 120 : imag matrix (base 256), entry (i,j), i< j, row-major
struct Lut { unsigned short v[NOUT]; };
constexpr Lut make_lut() {
    Lut t{};
    int k = 0;
    for (int i = 0; i < NZZ; ++i)
        for (int j = i; j < NZZ; ++j)
            t.v[k++] = (unsigned short)(i * 16 + j);
    for (int i = 0; i < NZZ; ++i)
        for (int j = i + 1; j < NZZ; ++j)
            t.v[k++] = (unsigned short)(256 + i * 16 + j);
    return t;
}
__constant__ Lut LUT = make_lut();

// ---- kernel: persistent waves, one row per wave per iteration ------------
__global__ void __launch_bounds__(THREADS)
biholo_k2_kernel(const float* __restrict__ x_re,
                 const float* __restrict__ x_im,
                 float* __restrict__ out,
                 int nrows)
{
    __shared__ float sm[WAVES_PER_BLOCK * 512];   // 2 x 16x16 f32 per wave

    const int tid   = threadIdx.x;
    const int lane  = tid & 31;
    const int wib   = tid >> 5;
    const int gwave = blockIdx.x * WAVES_PER_BLOCK + wib;
    const int nwave = (int)gridDim.x * WAVES_PER_BLOCK;

    // ---- loop-invariant per-lane state ----------------------------------
    constexpr unsigned long long I5P = pack3(I5T);
    constexpr unsigned long long J5P = pack3(J5T);

    const int l  = lane & 15;
    const int lc = (l < NZZ) ? l : 0;                  // clamp lane 15
    const int i5 = (int)((I5P >> (3 * lc)) & 7ull);
    const int j5 = (int)((J5P >> (3 * lc)) & 7ull);
    const bool act = (lane < NZZ);                     // lanes 15..31 = zero pad

    float* smw = &sm[wib * 512];
    const int n  = lane & 15;
    const int mh = (lane >> 4) << 3;                   // D: vgpr r -> M = mh + r

    // LUT gather addresses, hoisted out of the row loop (8 per lane)
    int roff[8];
#pragma unroll
    for (int t = 0; t < 8; ++t) {
        const int k = t * 32 + lane;
        roff[t] = (k < NOUT) ? (int)LUT.v[k] : 0;
    }

    // ---- grid-stride row loop (uniform per wave -> EXEC stays all-1s) ----
    for (int row = gwave; row < nrows; row += nwave) {
        const size_t base = (size_t)row * DIM5;        // same 1-2 cachelines per wave
        const float ar = __builtin_nontemporal_load(&x_re[base + i5]);
        const float ai = __builtin_nontemporal_load(&x_im[base + i5]);
        const float br = __builtin_nontemporal_load(&x_re[base + j5]);
        const float bi = __builtin_nontemporal_load(&x_im[base + j5]);

        float zr = ar * br - ai * bi;                  // Re(zz_l)
        float zi = ar * bi + ai * br;                  // Im(zz_l)
        zr = act ? zr : 0.0f;
        zi = act ? zi : 0.0f;

        // Rank-2 WMMAs: D_re = Re*Re^T + Im*Im^T ; D_im = Im*Re^T - Re*Im^T
        // A 16x4: vgpr0 = K0 (lanes 0-15) / K2 (lanes 16-31, zeroed), vgpr1 = K1/K3.
        // B 4x16 mirrors that striping.
        v2f a_re; a_re.x = zr;  a_re.y = zi;
        v2f a_im; a_im.x = zi;  a_im.y = -zr;
        v2f bb;   bb.x   = zr;  bb.y   = zi;

        v8f cz = {0.f, 0.f, 0.f, 0.f, 0.f, 0.f, 0.f, 0.f};
        v8f dre = __builtin_amdgcn_wmma_f32_16x16x4_f32(
            false, a_re, false, bb, (short)0, cz, false, false);
        v8f dim = __builtin_amdgcn_wmma_f32_16x16x4_f32(
            false, a_im, false, bb, (short)0, cz, false, false);

        // Spill D matrices to wave-private LDS in (M,N) order.
        // Same-wave LDS ops are in-order (DScnt) -> no workgroup barrier needed.
#pragma unroll
        for (int r = 0; r < 8; ++r) {
            smw[(mh + r) * 16 + n]       = dre[r];
            smw[256 + (mh + r) * 16 + n] = dim[r];
        }

        // Coalesced NT writeback: 32 consecutive lanes -> 32 consecutive floats
        const size_t obase = (size_t)row * NOUT;
#pragma unroll
        for (int t = 0; t < 8; ++t) {
            const int k = t * 32 + lane;
            const float v = smw[roff[t]];
            if (k < NOUT) __builtin_nontemporal_store(v, &out[obase + k]);
        }
    }
}

// ---- launcher ------------------------------------------------------------
extern "C" void kernel_launch(void* const* d_in, const int* in_sizes, int n_in,
                              void* d_out, int out_size, void* d_ws, size_t ws_size,
                              hipStream_t stream) {
    const float* x_re = (const float*)d_in[0];
    const float* x_im = (const float*)d_in[1];
    float* out = (float*)d_out;

    const int nrows = in_sizes[0] / DIM5;             // 1,000,000
    int blocks = (nrows + WAVES_PER_BLOCK - 1) / WAVES_PER_BLOCK;
    if (blocks > GRID_BLOCKS) blocks = GRID_BLOCKS;   // persistent waves

    hipLaunchKernelGGL(biholo_k2_kernel, dim3(blocks), dim3(THREADS), 0, stream,
                       x_re, x_im, out, nrows);
}